// PE_Flow3d_dense_heavy2_align_88201448391155
// MI455X (gfx1250) — compile-verified
//
#include <hip/hip_runtime.h>
#include <hip/hip_bf16.h>
#include <stdint.h>

// PointNet++-style set abstraction for MI455X (gfx1250, wave32).
// Kernel 1: ball query via ascending ballot-scan with early exit (first-16-by-index).
// Kernel 2: gather + 2-layer MLP on v_wmma_f32_16x16x4_f32 (fp32 matrix pipe) + neighbor max.
//           Weights/biases staged zero-padded in LDS once per block (no divergent
//           masked loads, no per-wave redundant weight fetches).

typedef __attribute__((ext_vector_type(2))) float v2f;
typedef __attribute__((ext_vector_type(8))) float v8f;

#define Bb 4
#define Nn 16384
#define Mm 4096
#define Kk 16
#define RADIUS2 0.25f

// ---------------- Kernel 1: ball query ----------------
// One wave (32 lanes) per centroid. Scan points in ascending index order,
// 32 at a time; ballot -> in-radius mask; popcount-prefix rank appends the
// hits in index order; early exit once 16 collected. Pad with first hit.
__global__ __launch_bounds__(256) void ball_query_kernel(
    const float* __restrict__ xyz, const int* __restrict__ fps_idx,
    int* __restrict__ grp)
{
    const int wave = (blockIdx.x * blockDim.x + threadIdx.x) >> 5;
    const int lane = threadIdx.x & 31;
    if (wave >= Bb * Mm) return;
    const int b = wave / Mm;
    const int m = wave - b * Mm;

    const int cidx = fps_idx[b * Mm + m];
    const float cx = xyz[(b * Nn + cidx) * 3 + 0];
    const float cy = xyz[(b * Nn + cidx) * 3 + 1];
    const float cz = xyz[(b * Nn + cidx) * 3 + 2];

    int collected = 0;
    int first_hit = -1;
    int* __restrict__ out = grp + wave * Kk;

    for (int base = 0; base < Nn; base += 32) {
        const int n = base + lane;                   // Nn % 32 == 0, always in range
        const float dx = xyz[(b * Nn + n) * 3 + 0] - cx;
        const float dy = xyz[(b * Nn + n) * 3 + 1] - cy;
        const float dz = xyz[(b * Nn + n) * 3 + 2] - cz;
        const float d2 = dx * dx + dy * dy + dz * dz;
        const bool pred = d2 <= RADIUS2;
        const unsigned mask = (unsigned)__ballot(pred);   // wave32: low 32 bits
        if (mask) {
            if (first_hit < 0) first_hit = base + __ffs(mask) - 1;
            const int rank = __popc(mask & ((1u << lane) - 1u));
            const int pos = collected + rank;
            if (pred && pos < Kk) out[pos] = n;
            collected += __popc(mask);
            if (collected >= Kk) break;              // early exit: first 16 by index found
        }
    }
    if (collected < Kk) {
        const int fill = (first_hit < 0) ? 0 : first_hit;   // matches ref padding semantics
        if (lane < Kk && lane >= collected) out[lane] = fill;
    }
}

// ---------------- Kernel 2: gather + WMMA MLP + max ----------------
// One wave per centroid: its 16 neighbors are the 16 rows (M-dim) of a
// 16x16x4 fp32 WMMA tile. A-layout: lane L<16 holds row L, K-bit0=vgpr,
// K-bit1=lane-half. C/D: reg r, lane l -> (M = r + 8*(l>=16), N = l%16).
__global__ __launch_bounds__(128) void group_mlp_wmma_kernel(
    const float* __restrict__ xyz, const float* __restrict__ feat,
    const int* __restrict__ fps_idx, const int* __restrict__ grp,
    const float* __restrict__ W1, const float* __restrict__ b1,
    const float* __restrict__ W2, const float* __restrict__ b2,
    float* __restrict__ out)
{
    // Weight staging (shared by all 4 waves in the block). Row stride 33 so the
    // two lane-halves (rows k and k+2, 32-dword base stride) hit disjoint banks.
    __shared__ float wbuf1[8][33];      // W1 zero-padded 6x32 -> 8x32
    __shared__ float wbuf2[32][33];     // W2 32x32
    __shared__ float bias1[32];
    __shared__ float bias2[32];
    __shared__ float hbuf[4][16][9];    // per-wave input rows, padded stride 9
    __shared__ float h1buf[4][16][33];  // per-wave h1 (16x32), padded stride 33

    const int tid   = threadIdx.x;
    const int wslot = tid >> 5;
    const int lane  = tid & 31;
    const int L     = lane & 15;        // row (A) / column (B,C,D) within tile
    const int half  = lane >> 4;        // K/lane-half selector
    const int wave  = blockIdx.x * 4 + wslot;
    const int b = wave / Mm;
    const int m = wave - b * Mm;

    // --- cooperative weight/bias staging (uniform, no divergence) ---
#pragma unroll
    for (int i = tid; i < 8 * 32; i += 128) {
        const int k = i >> 5, n = i & 31;
        wbuf1[k][n] = (k < 6) ? W1[k * 32 + n] : 0.f;   // zero-pad K 6 -> 8
    }
#pragma unroll
    for (int i = tid; i < 32 * 32; i += 128) {
        wbuf2[i >> 5][i & 31] = W2[i];
    }
    if (tid < 32) {
        bias1[tid] = b1[tid];
        bias2[tid] = b2[tid];
    }

    // --- gather: lanes 0..15 build one 6-wide input row each (pad to 8) ---
    if (half == 0) {
        const int cidx = fps_idx[b * Mm + m];
        const float cx = xyz[(b * Nn + cidx) * 3 + 0];
        const float cy = xyz[(b * Nn + cidx) * 3 + 1];
        const float cz = xyz[(b * Nn + cidx) * 3 + 2];
        const int sidx = grp[wave * Kk + L];
        const float* p = xyz  + (size_t)(b * Nn + sidx) * 3;
        const float* f = feat + (size_t)(b * Nn + sidx) * 3;
        hbuf[wslot][L][0] = p[0] - cx;
        hbuf[wslot][L][1] = p[1] - cy;
        hbuf[wslot][L][2] = p[2] - cz;
        hbuf[wslot][L][3] = f[0];
        hbuf[wslot][L][4] = f[1];
        hbuf[wslot][L][5] = f[2];
        hbuf[wslot][L][6] = 0.f;        // K padding (6 -> 8)
        hbuf[wslot][L][7] = 0.f;
    }
    __syncthreads();

    // --- layer 1: (16x8) @ (8x32), 2 K-chunks x 2 N-tiles = 4 WMMAs ---
    v2f a0, a1;
    a0.x = hbuf[wslot][L][0 + 2 * half];
    a0.y = hbuf[wslot][L][1 + 2 * half];
    a1.x = hbuf[wslot][L][4 + 2 * half];
    a1.y = hbuf[wslot][L][5 + 2 * half];   // half==1 -> rows 6,7 (zero pad)

    v8f acc1[2];
#pragma unroll
    for (int nt = 0; nt < 2; ++nt) {
        v8f c = {};
        const int k0 = 2 * half;           // K-chunk 0: rows 0..3
        v2f bw;
        bw.x = wbuf1[k0 + 0][nt * 16 + L];
        bw.y = wbuf1[k0 + 1][nt * 16 + L];
        c = __builtin_amdgcn_wmma_f32_16x16x4_f32(false, a0, false, bw,
                                                  (short)0, c, false, false);
        const int k4 = 4 + 2 * half;       // K-chunk 1: rows 4..7 (6,7 zero-padded)
        v2f bw2;
        bw2.x = wbuf1[k4 + 0][nt * 16 + L];
        bw2.y = wbuf1[k4 + 1][nt * 16 + L];
        c = __builtin_amdgcn_wmma_f32_16x16x4_f32(false, a1, false, bw2,
                                                  (short)0, c, false, false);
        acc1[nt] = c;
    }

    // --- bias + leaky(0.1), spill h1 (16x32) to LDS for the A-layout re-read ---
#pragma unroll
    for (int nt = 0; nt < 2; ++nt) {
        const float bias = bias1[nt * 16 + L];
#pragma unroll
        for (int r = 0; r < 8; ++r) {
            float x = acc1[nt][r] + bias;
            x = (x >= 0.f) ? x : 0.1f * x;
            h1buf[wslot][r + 8 * half][nt * 16 + L] = x;  // [M-row][N-col]
        }
    }
    __syncthreads();

    // --- layer 2: (16x32) @ (32x32), 8 K-chunks x 2 N-tiles = 16 WMMAs ---
    v8f acc2[2];
    acc2[0] = (v8f){};
    acc2[1] = (v8f){};
#pragma unroll
    for (int kc = 0; kc < 32; kc += 4) {
        v2f a;
        a.x = h1buf[wslot][L][kc + 2 * half + 0];
        a.y = h1buf[wslot][L][kc + 2 * half + 1];
#pragma unroll
        for (int nt = 0; nt < 2; ++nt) {
            v2f bw;
            bw.x = wbuf2[kc + 2 * half + 0][nt * 16 + L];
            bw.y = wbuf2[kc + 2 * half + 1][nt * 16 + L];
            acc2[nt] = __builtin_amdgcn_wmma_f32_16x16x4_f32(false, a, false, bw,
                                                             (short)0, acc2[nt],
                                                             false, false);
        }
    }

    // --- bias + leaky + max over the 16 neighbor rows ---
#pragma unroll
    for (int nt = 0; nt < 2; ++nt) {
        const float bias = bias2[nt * 16 + L];
        float mx = -3.402823466e38f;
#pragma unroll
        for (int r = 0; r < 8; ++r) {      // rows half*8 .. half*8+7 for column nt*16+L
            float x = acc2[nt][r] + bias;
            x = (x >= 0.f) ? x : 0.1f * x;
            mx = fmaxf(mx, x);
        }
        const float other = __shfl_xor(mx, 16, 32);  // merge the two lane-halves
        mx = fmaxf(mx, other);
        if (half == 0) out[(size_t)(b * Mm + m) * 32 + nt * 16 + L] = mx;
    }
}

extern "C" void kernel_launch(void* const* d_in, const int* in_sizes, int n_in,
                              void* d_out, int out_size, void* d_ws, size_t ws_size,
                              hipStream_t stream) {
    const float* xyz     = (const float*)d_in[0];
    const float* feat    = (const float*)d_in[1];
    const int*   fps_idx = (const int*)d_in[2];
    const float* W1      = (const float*)d_in[3];
    const float* b1      = (const float*)d_in[4];
    const float* W2      = (const float*)d_in[5];
    const float* b2      = (const float*)d_in[6];
    float* out = (float*)d_out;
    int*   grp = (int*)d_ws;                      // B*M*16 ints = 1 MB scratch

    // 1 wave per centroid, 8 waves per block
    const int n_centroids = Bb * Mm;              // 16384
    ball_query_kernel<<<(n_centroids * 32) / 256, 256, 0, stream>>>(xyz, fps_idx, grp);

    // 1 wave per centroid, 4 waves per block
    group_mlp_wmma_kernel<<<n_centroids / 4, 128, 0, stream>>>(
        xyz, feat, fps_idx, grp, W1, b1, W2, b2, out);
}